// S4DKernel_11708080849231
// MI455X (gfx1250) — compile-verified
//
#include <hip/hip_runtime.h>
#include <math.h>

// S4D Vandermonde kernel for MI455X (gfx1250), wave32 + WMMA.
//
// out[h, l] = 2 * Re( sum_n Cd[h,n] * exp(dtA[h,n] * l) ),  l in [0, L)
//
// Factorization: l = l0 + 16*l1  =>  z^l = z^l0 * (z^16)^l1, z = exp(dtA).
// Per head h this becomes a real GEMM:
//   A'[16 x 64]  = [ Re(2*Cd*z^l0) | Im(2*Cd*z^l0) ]          (l0 rows, n cols)
//   B'[64 x 128] = [ Re(W^l1) ; -Im(W^l1) ],  W = z^16        (n rows, l1 cols)
//   out_tile[16 x 128] = A' @ B'
// mapped to V_WMMA_F32_16X16X4_F32: K=64 -> 16 WMMA steps, N-tile loop of L/256.
//
// Transcendental minimization (setup dominates the 8-iteration tile loop):
//  - stage 1: ONE expf + sincosf per n gives z; 8 complex squarings give
//    z^(2^j), j=0..8, in LDS (float2 -> ds_load_b64 broadcasts).
//  - stage 2: per lane, 2Cd*z^m and z^(16m) via <=4 conditional complex
//    multiplies each (bits of m select from the power table); tile step z^256
//    read directly. No libm calls at all here.
//  - tile loop: B state stored with Im pre-negated so fragments need no xor;
//    advance is one complex multiply per element (W^(l1+16) = W^l1 * z^256).

typedef __attribute__((ext_vector_type(2))) float v2f;
typedef __attribute__((ext_vector_type(8))) float v8f;

__global__ __launch_bounds__(32) void s4d_wmma_kernel(
    const float* __restrict__ log_dt,      // (H,)
    const float* __restrict__ Cri,         // (H, 32, 2)
    const float* __restrict__ log_A_real,  // (H, 32)
    const float* __restrict__ A_imag,      // (H, 32)
    float* __restrict__ out,               // (H, L)
    int L)
{
    const int h    = blockIdx.x;
    const int lane = threadIdx.x;          // 0..31, one full wave32 per head

    __shared__ float2 s_pow[9][32];        // z^(2^j) per n, j = 0..8
    __shared__ float2 s_cd[32];            // 2*Cd per n

    // ---- Stage 1: lane n computes z^(2^j) table and 2*Cd_n ----
    {
        const int n   = lane;
        const int idx = h * 32 + n;
        const float dt  = expf(log_dt[h]);
        const float are = -expf(log_A_real[idx]);   // Re(A)
        const float aim = A_imag[idx];              // Im(A)
        const float dre = are * dt;                 // Re(dtA)
        const float dim = aim * dt;                 // Im(dtA)

        float sn, cs;
        sincosf(dim, &sn, &cs);
        const float ex  = expf(dre);
        const float z1r = ex * cs;                  // z = exp(dtA)
        const float z1i = ex * sn;

        // 2*Cd = 2*C*(z-1)/A
        const float zr  = z1r - 1.0f;
        const float zi  = z1i;
        const float den = are * are + aim * aim;
        const float tr  = (zr * are + zi * aim) / den;
        const float ti  = (zi * are - zr * aim) / den;
        const float Cr  = Cri[2 * idx + 0];
        const float Ci  = Cri[2 * idx + 1];
        s_cd[n] = make_float2(2.0f * (Cr * tr - Ci * ti),
                              2.0f * (Cr * ti + Ci * tr));

        // power table by repeated squaring: z, z^2, z^4, ..., z^256
        float pr = z1r, pi = z1i;
        s_pow[0][n] = make_float2(pr, pi);
#pragma unroll
        for (int j = 1; j <= 8; ++j) {
            const float nr = pr * pr - pi * pi;
            const float ni = 2.0f * pr * pi;
            pr = nr; pi = ni;
            s_pow[j][n] = make_float2(pr, pi);
        }
    }
    __syncthreads();

    // Fragment index mapping for V_WMMA_F32_16X16X4_F32 (wave32):
    //   A 16x4 : lanes 0-15 -> M=lane,   K = kb+e (kb=0); lanes 16-31 -> kb=2
    //   B 4x16 : lanes 0-15 -> N=lane,   K = kb+e;        lanes 16-31 -> kb=2
    //   D 16x16: VGPR r: lanes 0-15 (M=r, N=lane), lanes 16-31 (M=r+8, N=lane-16)
    const int m  = lane & 15;
    const int kb = (lane >> 4) << 1;

    v2f afrag[16];                 // A: [f]=Re half (k<32), [f+8]=Im half
    v2f bR[8], bNI[8];             // B state: Re(W^l1) and -Im(W^l1)
    v2f wR[8], wI[8];              // tile step z^256

#pragma unroll
    for (int f = 0; f < 8; ++f) {
#pragma unroll
        for (int e = 0; e < 2; ++e) {
            const int n = 4 * f + kb + e;

            // A-side accumulator starts at 2Cd; multiply by z^(2^j) per bit of m
            const float2 cd = s_cd[n];
            float ar = cd.x, ai = cd.y;

            // B-side: z^(16m) = prod over bits j of m of z^(2^(j+4))
            float brv, biv;
            {
                const float2 p = s_pow[4][n];
                const bool b0 = (m & 1) != 0;
                brv = b0 ? p.x : 1.0f;
                biv = b0 ? p.y : 0.0f;
            }
#pragma unroll
            for (int j = 0; j < 4; ++j) {
                const bool bit = ((m >> j) & 1) != 0;
                const float2 pa = s_pow[j][n];
                const float qr = bit ? pa.x : 1.0f;
                const float qi = bit ? pa.y : 0.0f;
                const float t0 = ar * qr - ai * qi;
                const float t1 = ar * qi + ai * qr;
                ar = t0; ai = t1;
                if (j >= 1) {
                    const float2 pb = s_pow[j + 4][n];
                    const float ur = bit ? pb.x : 1.0f;
                    const float ui = bit ? pb.y : 0.0f;
                    const float s0 = brv * ur - biv * ui;
                    const float s1 = brv * ui + biv * ur;
                    brv = s0; biv = s1;
                }
            }

            afrag[f][e]     = ar;              // Re(2Cd * z^m)
            afrag[f + 8][e] = ai;              // Im(2Cd * z^m)
            bR[f][e]  = brv;                   //  Re(W^m)
            bNI[f][e] = -biv;                  // -Im(W^m)  (pre-negated state)
            const float2 st = s_pow[8][n];     // z^256 = W^16
            wR[f][e] = st.x;
            wI[f][e] = st.y;
        }
    }

    // ---- Tile loop: 16-step WMMA chain, b128 stores, complex-mult advance ----
    const int ntiles = L >> 8;                       // L / (16*16)
    float* __restrict__ outp = out + (size_t)h * (size_t)L;
    const int rowoff = (lane >> 4) * 8;              // D-matrix M offset per half-wave

    for (int t = 0; t < ntiles; ++t) {
        v8f acc = {0.f, 0.f, 0.f, 0.f, 0.f, 0.f, 0.f, 0.f};
#pragma unroll
        for (int f = 0; f < 8; ++f) {
            acc = __builtin_amdgcn_wmma_f32_16x16x4_f32(
                false, afrag[f], false, bR[f], (short)0, acc, false, false);
        }
#pragma unroll
        for (int f = 0; f < 8; ++f) {
            acc = __builtin_amdgcn_wmma_f32_16x16x4_f32(
                false, afrag[f + 8], false, bNI[f], (short)0, acc, false, false);
        }

        // D element (M=r+rowoff, N=l1) -> out index l = (r+rowoff) + 16*l1:
        // the 8 accumulators per lane are 8 consecutive floats.
        const int l1   = 16 * t + m;
        const int base = rowoff + 16 * l1;
        float4 lo, hi;
        lo.x = acc[0]; lo.y = acc[1]; lo.z = acc[2]; lo.w = acc[3];
        hi.x = acc[4]; hi.y = acc[5]; hi.z = acc[6]; hi.w = acc[7];
        *(float4*)(outp + base)     = lo;            // global_store_b128
        *(float4*)(outp + base + 4) = hi;            // global_store_b128

        // advance: W^(l1+16) = W^l1 * z^256, on (Re, -Im) state => no xors:
        //   br'  = br*wr + nbi*wi
        //   nbi' = nbi*wr - br*wi
        if (t + 1 < ntiles) {
#pragma unroll
            for (int f = 0; f < 8; ++f) {
#pragma unroll
                for (int e = 0; e < 2; ++e) {
                    const float r0 = bR[f][e], n0 = bNI[f][e];
                    bR[f][e]  = r0 * wR[f][e] + n0 * wI[f][e];
                    bNI[f][e] = n0 * wR[f][e] - r0 * wI[f][e];
                }
            }
        }
    }
}

extern "C" void kernel_launch(void* const* d_in, const int* in_sizes, int n_in,
                              void* d_out, int out_size, void* d_ws, size_t ws_size,
                              hipStream_t stream) {
    const float* log_dt     = (const float*)d_in[0];
    const float* C          = (const float*)d_in[1];
    const float* log_A_real = (const float*)d_in[2];
    const float* A_imag     = (const float*)d_in[3];
    // d_in[4] is L as a device-side scalar; derive it host-side instead
    // (graph-capture-safe): L = out_size / H.
    const int H = in_sizes[0];
    const int L = out_size / H;                      // 2048 for the reference
    float* out = (float*)d_out;

    dim3 grid(H), block(32);
    hipLaunchKernelGGL(s4d_wmma_kernel, grid, block, 0, stream,
                       log_dt, C, log_A_real, A_imag, out, L);
}